// POS_predictor_20804821582148
// MI455X (gfx1250) — compile-verified
//
#include <hip/hip_runtime.h>
#include <hip/hip_bf16.h>

// ---------------- problem constants (from setup_inputs) ----------------
constexpr int kB  = 64;          // batch
constexpr int kT  = 64;          // sequence length
constexpr int kC  = 16;          // chars per word
constexpr int kE  = 512;         // word embedding dim
constexpr int kH  = 512;         // word GRU hidden
constexpr int kCE = 128;         // char embedding dim
constexpr int kCH = 256;         // char GRU hidden
constexpr int kNC = 18;          // classes
constexpr int kBT = kB * kT;     // 4096 words
constexpr int kKW = kE + kCH;    // 768, word-GRU input dim
constexpr int k3H = 3 * kH;      // 1536

// padded LDS row strides (floats): bank-conflict-free (stride%64==4)
// and 16B-aligned (stride*4 % 16 == 0) for b128 staging stores.
constexpr int kSIH = kCE + 4;    // 132
constexpr int kSHH = kCH + 4;    // 260
constexpr int kLdsIh = 48 * kSIH;            // 6336 floats
constexpr int kLdsHh = 48 * kSHH;            // 12480 floats
constexpr int kLdsBytes = (kLdsIh + kLdsHh) * 4;   // 75264 B dynamic LDS

typedef __attribute__((ext_vector_type(2))) float v2f;
typedef __attribute__((ext_vector_type(8))) float v8f;
typedef int vint4 __attribute__((vector_size(16)));   // matches builtin param type

__device__ __forceinline__ v2f ld2(const float* p) {
    return *reinterpret_cast<const v2f*>(p);
}

__device__ __forceinline__ float sigm(float x) {
    return 1.0f / (1.0f + __expf(-x));
}

// D = A(16x4 f32) * B(4x16 f32) + C, fp32 WMMA (codegen-confirmed round 1)
__device__ __forceinline__ v8f wmma4(v2f a, v2f b, v8f c) {
    return __builtin_amdgcn_wmma_f32_16x16x4_f32(
        false, a, false, b, (short)0, c, false, false);
}

// ---- global -> LDS staging: async (ASYNCcnt-tracked) when available ----
#if __has_builtin(__builtin_amdgcn_global_load_async_to_lds_b128)
#define HAS_ASYNC_LDS 1
#endif

__device__ __forceinline__ void cp16_to_lds(const float* g, float* l) {
#ifdef HAS_ASYNC_LDS
    __builtin_amdgcn_global_load_async_to_lds_b128(
        (__attribute__((address_space(1))) vint4*)(float*)g,
        (__attribute__((address_space(3))) vint4*)l, 0, 0);
#else
    const float4 v = *(const float4*)g;
    *(float4*)l = v;
#endif
}

__device__ __forceinline__ void stage_wait() {
#ifdef HAS_ASYNC_LDS
#if __has_builtin(__builtin_amdgcn_s_wait_asynccnt)
    __builtin_amdgcn_s_wait_asynccnt(0);
#else
    asm volatile("s_wait_asynccnt 0x0" ::: "memory");
#endif
#endif
    __syncthreads();
}

// ---------------------------------------------------------------------
// Char GRU step. Block = 8 waves covering 512 rows x one 16-col tile.
// The column tile's weight rows (3 gates of Wih and Whh) are staged in
// dynamic LDS once per block (async-to-LDS path); each wave computes 4
// row-tiles (M=64), so each LDS B-fragment feeds 4 WMMAs.
//   r,z fused across both GEMMs; n gate split (r scales only gh_n).
// grid = (BT/512) * (CH/16) = 128 blocks of 256 threads.
// ---------------------------------------------------------------------
__global__ __launch_bounds__(256) void char_step_kernel(
    const int* __restrict__ chars, const float* __restrict__ cemb,
    const float* __restrict__ Wih, const float* __restrict__ Whh,
    const float* __restrict__ bih, const float* __restrict__ bhh,
    const float* __restrict__ h_in, float* __restrict__ h_out,
    int t, int first)
{
    extern __shared__ float sB[];           // [48][kSIH] then [48][kSHH]
    float* sIh = sB;
    float* sHh = sB + kLdsIh;

    const int wid  = threadIdx.x >> 5;
    const int lane = threadIdx.x & 31;
    const int lm   = lane & 15;
    const int half = lane >> 4;
    const int n0   = (blockIdx.x & 15) << 4;           // 16 column tiles
    const int m0   = ((blockIdx.x >> 4) << 9) + (wid << 6);  // 512 rows/block, 64/wave

    // ---- stage weight tile into LDS (row fr = gate*16 + r) ----
    {
        const int tid = threadIdx.x;
        #pragma unroll
        for (int it = 0; it < 6; ++it) {                 // 1536 float4 of Wih
            const int i  = tid + it * 256;
            const int fr = i >> 5, kk = (i & 31) << 2;
            const int g = fr >> 4, r = fr & 15;
            cp16_to_lds(Wih + (long)(g * kCH + n0 + r) * kCE + kk,
                        sIh + fr * kSIH + kk);
        }
        #pragma unroll
        for (int it = 0; it < 12; ++it) {                // 3072 float4 of Whh
            const int i  = tid + it * 256;
            const int fr = i >> 6, kk = (i & 63) << 2;
            const int g = fr >> 4, r = fr & 15;
            cp16_to_lds(Whh + (long)(g * kCH + n0 + r) * kCH + kk,
                        sHh + fr * kSHH + kk);
        }
    }
    stage_wait();

    v8f accr[4], accz[4], gxn[4], ghn[4];
    #pragma unroll
    for (int j = 0; j < 4; ++j) { accr[j] = {}; accz[j] = {}; gxn[j] = {}; ghn[j] = {}; }

    // A row pointers: 4 row-tiles, lane supplies row (tile_m0 + lm)
    const float* ae[4];
    #pragma unroll
    for (int j = 0; j < 4; ++j) {
        const int ci = chars[(m0 + j * 16 + lm) * kC + t];
        ae[j] = cemb + (long)ci * kCE;
    }

    // ---- input-projection GEMM (K = CE), B from LDS ----
    const float* br = sIh + (0 * 16 + lm) * kSIH;
    const float* bz = sIh + (1 * 16 + lm) * kSIH;
    const float* bn = sIh + (2 * 16 + lm) * kSIH;
    #pragma unroll 2
    for (int k = 0; k < kCE; k += 4) {
        const int o = k + 2 * half;
        v2f fbr = ld2(br + o), fbz = ld2(bz + o), fbn = ld2(bn + o);
        #pragma unroll
        for (int j = 0; j < 4; ++j) {
            v2f a = ld2(ae[j] + o);
            accr[j] = wmma4(a, fbr, accr[j]);
            accz[j] = wmma4(a, fbz, accz[j]);
            gxn[j]  = wmma4(a, fbn, gxn[j]);
        }
    }

    // ---- hidden GEMM (K = CH), skipped on step 0 (h0 == 0) ----
    if (!first) {
        const float* hr = sHh + (0 * 16 + lm) * kSHH;
        const float* hz = sHh + (1 * 16 + lm) * kSHH;
        const float* hn = sHh + (2 * 16 + lm) * kSHH;
        #pragma unroll 2
        for (int k = 0; k < kCH; k += 4) {
            const int o = k + 2 * half;
            v2f fbr = ld2(hr + o), fbz = ld2(hz + o), fbn = ld2(hn + o);
            #pragma unroll
            for (int j = 0; j < 4; ++j) {
                v2f a = ld2(h_in + (long)(m0 + j * 16 + lm) * kCH + o);
                accr[j] = wmma4(a, fbr, accr[j]);
                accz[j] = wmma4(a, fbz, accz[j]);
                ghn[j]  = wmma4(a, fbn, ghn[j]);
            }
        }
    }

    // ---- gates + state update ----
    const int ncol = n0 + lm;
    const float brr = bih[ncol] + bhh[ncol];
    const float bzz = bih[kCH + ncol] + bhh[kCH + ncol];
    const float bnx = bih[2 * kCH + ncol];
    const float bnh = bhh[2 * kCH + ncol];
    #pragma unroll
    for (int j = 0; j < 4; ++j) {
        #pragma unroll
        for (int v = 0; v < 8; ++v) {
            const int m = m0 + j * 16 + v + 8 * half;
            float r  = sigm(accr[j][v] + brr);
            float z  = sigm(accz[j][v] + bzz);
            float nn = tanhf(gxn[j][v] + bnx + r * (ghn[j][v] + bnh));
            float ho = first ? 0.0f : h_in[(long)m * kCH + ncol];
            h_out[(long)m * kCH + ncol] = (1.0f - z) * nn + z * ho;
        }
    }
}

// ---------------------------------------------------------------------
// Word-GRU input projection: gxw[m,:] = [word_emb[x[m]], char_h[m]] @ Wih.T + bih
// Each wave: 4 row-tiles x 2 col-tiles (8 accumulators), K = 768.
// tiles = (4096/64) * (1536/32) = 3072 waves -> 384 blocks.
// ---------------------------------------------------------------------
__global__ __launch_bounds__(256) void gxw_kernel(
    const int* __restrict__ x, const float* __restrict__ wemb,
    const float* __restrict__ char_h, const float* __restrict__ Wih,
    const float* __restrict__ bih, float* __restrict__ gxw)
{
    const int wid  = threadIdx.x >> 5;
    const int tile = blockIdx.x * 8 + wid;
    const int lane = threadIdx.x & 31;
    const int lm   = lane & 15;
    const int half = lane >> 4;
    const int m0   = (tile / 48) << 6;   // 64 rows per wave
    const int n0   = (tile % 48) << 5;   // 32 cols per wave

    const float* ae[4];
    const float* ac[4];
    #pragma unroll
    for (int j = 0; j < 4; ++j) {
        const int row = m0 + j * 16 + lm;
        ae[j] = wemb + (long)x[row] * kE;
        ac[j] = char_h + (long)row * kCH;
    }
    const float* bw0 = Wih + (long)(n0 + lm) * kKW;
    const float* bw1 = Wih + (long)(n0 + 16 + lm) * kKW;

    v8f acc[4][2];
    #pragma unroll
    for (int j = 0; j < 4; ++j) { acc[j][0] = {}; acc[j][1] = {}; }

    #pragma unroll 2
    for (int k = 0; k < kKW; k += 4) {
        const int o = k + 2 * half;
        v2f b0 = ld2(bw0 + o), b1 = ld2(bw1 + o);
        #pragma unroll
        for (int j = 0; j < 4; ++j) {
            v2f a = (k < kE) ? ld2(ae[j] + o) : ld2(ac[j] + (o - kE));
            acc[j][0] = wmma4(a, b0, acc[j][0]);
            acc[j][1] = wmma4(a, b1, acc[j][1]);
        }
    }

    #pragma unroll
    for (int n = 0; n < 2; ++n) {
        const int ncol = n0 + n * 16 + lm;
        const float bb = bih[ncol];
        #pragma unroll
        for (int j = 0; j < 4; ++j) {
            #pragma unroll
            for (int v = 0; v < 8; ++v) {
                const int m = m0 + j * 16 + v + 8 * half;
                gxw[(long)m * k3H + ncol] = acc[j][n][v] + bb;
            }
        }
    }
}

// ---------------------------------------------------------------------
// Word GRU step t: gh = h @ Whh.T (+bhh), gates, h update, store hs[b,t,:].
// One wave per 16x16 tile; 128 tiles -> 16 blocks (keeps per-step latency low).
// ---------------------------------------------------------------------
__global__ __launch_bounds__(256) void word_step_kernel(
    const float* __restrict__ gxw, const float* __restrict__ Whh,
    const float* __restrict__ bhh, const float* __restrict__ h_in,
    float* __restrict__ h_out, float* __restrict__ hs,
    int t, int first)
{
    const int wid  = threadIdx.x >> 5;
    const int tile = blockIdx.x * 8 + wid;
    const int lane = threadIdx.x & 31;
    const int lm   = lane & 15;
    const int half = lane >> 4;
    const int m0   = (tile >> 5) << 4;   // 32 column tiles (H/16)
    const int n0   = (tile & 31) << 4;

    v8f ar = {}, az = {}, an = {};
    if (!first) {
        const float* ah = h_in + (long)(m0 + lm) * kH;
        const float* wr = Whh + (long)(0 * kH + n0 + lm) * kH;
        const float* wz = Whh + (long)(1 * kH + n0 + lm) * kH;
        const float* wn = Whh + (long)(2 * kH + n0 + lm) * kH;
        #pragma unroll 4
        for (int k = 0; k < kH; k += 4) {
            const int o = k + 2 * half;
            v2f a = ld2(ah + o);
            ar = wmma4(a, ld2(wr + o), ar);
            az = wmma4(a, ld2(wz + o), az);
            an = wmma4(a, ld2(wn + o), an);
        }
    }

    const int ncol = n0 + lm;
    const float brr = bhh[ncol];
    const float bzz = bhh[kH + ncol];
    const float bnn = bhh[2 * kH + ncol];
    #pragma unroll
    for (int v = 0; v < 8; ++v) {
        const int brow = m0 + v + 8 * half;           // batch index
        const float* gx = gxw + ((long)brow * kT + t) * k3H;
        float r  = sigm(gx[ncol] + ar[v] + brr);
        float z  = sigm(gx[kH + ncol] + az[v] + bzz);
        float nn = tanhf(gx[2 * kH + ncol] + r * (an[v] + bnn));
        float ho = first ? 0.0f : h_in[(long)brow * kH + ncol];
        float hn = (1.0f - z) * nn + z * ho;
        h_out[(long)brow * kH + ncol] = hn;
        hs[((long)brow * kT + t) * kH + ncol] = hn;
    }
}

// ---------------------------------------------------------------------
// Classifier: out[bt, c] = hs[bt,:] . cls_W[c,:] + cls_b[c]   (N=18, tiny)
// ---------------------------------------------------------------------
__global__ __launch_bounds__(256) void cls_kernel(
    const float* __restrict__ hs, const float* __restrict__ W,
    const float* __restrict__ b, float* __restrict__ out)
{
    const int id = blockIdx.x * blockDim.x + threadIdx.x;
    if (id >= kBT * kNC) return;
    const int bt = id / kNC;
    const int c  = id % kNC;
    const float* hv = hs + (long)bt * kH;
    const float* wv = W + (long)c * kH;
    float s = b[c];
    #pragma unroll 8
    for (int k = 0; k < kH; ++k) s += hv[k] * wv[k];
    out[id] = s;
}

// ---------------------------------------------------------------------
extern "C" void kernel_launch(void* const* d_in, const int* in_sizes, int n_in,
                              void* d_out, int out_size, void* d_ws, size_t ws_size,
                              hipStream_t stream) {
    const int*   x     = (const int*)d_in[0];
    const int*   chars = (const int*)d_in[1];
    const float* wemb  = (const float*)d_in[2];
    const float* cemb  = (const float*)d_in[3];
    const float* cWih  = (const float*)d_in[4];
    const float* cWhh  = (const float*)d_in[5];
    const float* cbih  = (const float*)d_in[6];
    const float* cbhh  = (const float*)d_in[7];
    const float* gWih  = (const float*)d_in[8];
    const float* gWhh  = (const float*)d_in[9];
    const float* gbih  = (const float*)d_in[10];
    const float* gbhh  = (const float*)d_in[11];
    const float* clsW  = (const float*)d_in[12];
    const float* clsb  = (const float*)d_in[13];
    float* out = (float*)d_out;

    // workspace carve-out (floats): ~42 MB
    float* w   = (float*)d_ws;
    float* hc0 = w;  w += (size_t)kBT * kCH;     // 4 MB
    float* hc1 = w;  w += (size_t)kBT * kCH;     // 4 MB
    float* gxw = w;  w += (size_t)kBT * k3H;     // 24 MB
    float* hw0 = w;  w += (size_t)kB * kH;       // 128 KB
    float* hw1 = w;  w += (size_t)kB * kH;       // 128 KB
    float* hs  = w;  w += (size_t)kBT * kH;      // 8 MB

    // ---- char GRU: 16 sequential steps, ping-pong h buffers ----
    float* hc[2] = {hc0, hc1};
    for (int t = 0; t < kC; ++t) {
        // (BT/512)*(CH/16) = 128 blocks, 75 KB dynamic LDS each
        char_step_kernel<<<128, 256, kLdsBytes, stream>>>(
            chars, cemb, cWih, cWhh, cbih, cbhh,
            hc[t & 1], hc[(t + 1) & 1], t, (t == 0) ? 1 : 0);
    }
    float* char_h = hc[kC & 1];  // final hidden after 16 steps (== hc0)

    // ---- word GRU input projection (one big GEMM) ----
    gxw_kernel<<<384, 256, 0, stream>>>(x, wemb, char_h, gWih, gbih, gxw);

    // ---- word GRU: 64 sequential steps ----
    float* hw[2] = {hw0, hw1};
    for (int t = 0; t < kT; ++t) {
        word_step_kernel<<<16, 256, 0, stream>>>(
            gxw, gWhh, gbhh, hw[t & 1], hw[(t + 1) & 1], hs, t, (t == 0) ? 1 : 0);
    }

    // ---- classifier ----
    cls_kernel<<<(kBT * kNC + 255) / 256, 256, 0, stream>>>(hs, clsW, clsb, out);
}